// OrbitalAEVComputer_15307263443409
// MI455X (gfx1250) — compile-verified
//
#include <hip/hip_runtime.h>
#include <hip/hip_bf16.h>

typedef __attribute__((ext_vector_type(2))) float v2f;
typedef __attribute__((ext_vector_type(8))) float v8f;

#define NCONF 1024
#define NATOMS 64
#define NS 9
#define OUT_PER_ATOM 1744   // 144 (s) + 64 (radial) + 1536 (angular)
#define RAD_OFF 144
#define ANG_OFF 208
#define PI_F 3.14159265358979323846f

// One wave32 per (conf, atom). Fully fused: s_aev, radial_aev, angular_aev.
// angular outer products f_theta(16) x f_dist(16) computed with
// V_WMMA_F32_16X16X4_F32 (A col0 = f_theta, B row0 = f_dist, rest zero).
__global__ __launch_bounds__(256)
void orbital_aev_kernel(const float* __restrict__ coeff, float* __restrict__ out)
{
    const int tid  = blockIdx.x * blockDim.x + threadIdx.x;
    const int wid  = tid >> 5;              // wave id == (conf*NATOMS + atom)
    const int lane = threadIdx.x & 31;

    const float* __restrict__ cp = coeff + (size_t)wid * 21;
    float* __restrict__ ob = out + (size_t)wid * OUT_PER_ATOM;

    // ---------------- s_aev: 9 rows x 16 shifts, OShfS = linspace(-4,4,16) --
    #pragma unroll
    for (int k = 0; k < 5; ++k) {
        int idx = k * 32 + lane;
        if (idx < NS * 16) {
            int i = idx >> 4, j = idx & 15;
            float s   = cp[i];
            float shf = -4.0f + 8.0f * (float)j * (1.0f / 15.0f);
            float d   = s - shf;
            __builtin_nontemporal_store(__expf(-16.0f * d * d), ob + idx);
        }
    }

    // ---------------- p vectors: norms + unit vectors -----------------------
    float nrm[4], hx[4], hy[4], hz[4];
    #pragma unroll
    for (int v = 0; v < 4; ++v) {
        float x = cp[NS + 3 * v + 0];
        float y = cp[NS + 3 * v + 1];
        float z = cp[NS + 3 * v + 2];
        float n2 = x * x + y * y + z * z;
        nrm[v] = sqrtf(n2);
        bool zero = (fabsf(x) < 1e-12f) && (fabsf(y) < 1e-12f) && (fabsf(z) < 1e-12f);
        float inv = zero ? 0.0f : (1.0f / nrm[v]);
        hx[v] = x * inv; hy[v] = y * inv; hz[v] = z * inv;
    }

    // ---------------- radial_aev: 4 x 16, OShfR = linspace(0,4,16) ----------
    #pragma unroll
    for (int k = 0; k < 2; ++k) {
        int idx = k * 32 + lane;
        int v = idx >> 4, j = idx & 15;
        float shf = 4.0f * (float)j * (1.0f / 15.0f);
        float d   = nrm[v] - shf;
        __builtin_nontemporal_store(__expf(-8.0f * d * d), ob + RAD_OFF + idx);
    }

    // ---------------- angular_aev: 6 pairs x (16 theta x 16 dist) -----------
    const int   t       = lane & 15;
    const float theta_t = PI_F * (float)t * (1.0f / 15.0f);   // OShfTheta
    const float ct      = __cosf(theta_t);
    const float st      = __sinf(theta_t);
    const float shA     = 4.0f * (float)t * (1.0f / 15.0f);   // OShfA
    const bool  lo      = (lane < 16);

    const int IU[6] = {0, 0, 0, 1, 1, 2};
    const int JU[6] = {1, 2, 3, 2, 3, 3};

    #pragma unroll
    for (int p = 0; p < 6; ++p) {
        const int i = IU[p], j = JU[p];
        float c = hx[i] * hx[j] + hy[i] * hy[j] + hz[i] * hz[j];
        c = fminf(fmaxf(c, -0.9999f), 0.9999f);
        float s = sqrtf(fmaxf(1.0f - c * c, 0.0f));  // sin(acos(c)) >= 0
        // cos(angle - theta_t) = c*cos(theta_t) + s*sin(theta_t)
        float x  = 0.5f * (1.0f + c * ct + s * st);
        float x2 = x * x, x4 = x2 * x2, x8 = x4 * x4;
        float ftheta = 0.0078125f * x8;              // 2^(1-8) * x^8

        float avd   = 0.5f * (nrm[i] + nrm[j]);
        float dd    = avd - shA;
        float fdist = __expf(-8.0f * dd * dd);

        // Rank-1 outer product via WMMA: D[t][a] = ftheta[t] * fdist[a]
        // A (16x4): col K=0 = ftheta (lanes 0-15, VGPR0); all else zero.
        // B (4x16): row K=0 = fdist  (lanes 0-15, VGPR0); all else zero.
        v2f A, B;
        A.x = lo ? ftheta : 0.0f;  A.y = 0.0f;
        B.x = lo ? fdist  : 0.0f;  B.y = 0.0f;
        v8f C = {};
        v8f D = __builtin_amdgcn_wmma_f32_16x16x4_f32(
            /*neg_a=*/false, A, /*neg_b=*/false, B,
            /*c_mod=*/(short)0, C, /*reuse_a=*/false, /*reuse_b=*/false);

        // D layout: VGPR r holds row M=r (lanes 0-15) and M=r+8 (lanes 16-31).
        // Output flattening: pair*256 + t*16 + a  -> coalesced dword stores.
        float* __restrict__ op = ob + ANG_OFF + p * 256;
        const int N  = lane & 15;
        const int Mb = (lane >> 4) * 8;
        #pragma unroll
        for (int r = 0; r < 8; ++r)
            __builtin_nontemporal_store(D[r], op + (Mb + r) * 16 + N);
    }
}

extern "C" void kernel_launch(void* const* d_in, const int* in_sizes, int n_in,
                              void* d_out, int out_size, void* d_ws, size_t ws_size,
                              hipStream_t stream) {
    const float* coeff = (const float*)d_in[0];   // (1024, 64, 21) f32
    // d_in[1] = normalization_library (unused by reference output)
    // d_in[2] = species (unused by reference output)
    float* out = (float*)d_out;                   // (1024, 64, 1744) f32

    const int total_waves = NCONF * NATOMS;       // 65536
    dim3 block(256);                              // 8 waves per block
    dim3 grid(total_waves * 32 / 256);            // 8192 blocks, exact fit
    hipLaunchKernelGGL(orbital_aev_kernel, grid, block, 0, stream, coeff, out);
}